// QLSTM_65481071406610
// MI455X (gfx1250) — compile-verified
//
#include <hip/hip_runtime.h>
#include <hip/hip_bf16.h>
#include <math.h>

typedef __attribute__((ext_vector_type(2))) float v2f;
typedef __attribute__((ext_vector_type(8))) float v8f;

#define NQ      8
#define QDEPTH  2
#define DIMQ    256
#define TSTEPS  256
#define BATCH   128
#define DFEAT   512
#define NROWS   (TSTEPS * BATCH)   /* 32768 */

/* ---- workspace layout (bytes) ---- */
#define CONST_OFF 0                               /* 4 gates * 8 floats        */
#define WB_OFF    1024                            /* 256*16*2 floats = 32 KB   */
#define PRE_OFF   (WB_OFF + 32768)                /* NROWS*4 floats = 512 KB   */
#define HSEQ_OFF  (PRE_OFF + NROWS * 4 * 4)       /* NROWS floats = 128 KB     */
#define HFIN_OFF  (HSEQ_OFF + NROWS * 4)          /* 128 floats                */
#define CFIN_OFF  (HFIN_OFF + BATCH * 4)          /* 128 floats                */

/* =====================================================================
 * Kernel 1: per-gate reduced constants {C0, C1, C2, sum(W[0,512:]), b[0]}
 * One 256-thread block per gate. Statevector sim of U|0> and U|128> in LDS.
 * ===================================================================== */
__global__ void qlstm_gate_consts(const float* __restrict__ Wf, const float* __restrict__ bf, const float* __restrict__ Pf,
                                  const float* __restrict__ Wi, const float* __restrict__ bi, const float* __restrict__ Pi,
                                  const float* __restrict__ Wg, const float* __restrict__ bg, const float* __restrict__ Pg,
                                  const float* __restrict__ Wo, const float* __restrict__ bo, const float* __restrict__ Po,
                                  float* __restrict__ consts) {
    const int g = blockIdx.x;
    const float* W; const float* bv; const float* P;
    switch (g) {
        case 0:  W = Wf; bv = bf; P = Pf; break;
        case 1:  W = Wi; bv = bi; P = Pi; break;
        case 2:  W = Wg; bv = bg; P = Pg; break;
        default: W = Wo; bv = bo; P = Po; break;
    }
    const int k = threadIdx.x;   /* amplitude index 0..255 */

    __shared__ float sr0[DIMQ], si0[DIMQ], sr1[DIMQ], si1[DIMQ];

    /* col0 = U e_0 ; col1 = U e_128 */
    float re0 = (k == 0)   ? 1.0f : 0.0f, im0 = 0.0f;
    float re1 = (k == 128) ? 1.0f : 0.0f, im1 = 0.0f;

    for (int l = 0; l < QDEPTH; ++l) {
        /* single-qubit rotation layer: qubit w acts on bit p = 7-w */
        for (int w = 0; w < NQ; ++w) {
            const int p = 7 - w;
            const float phi = P[(l * NQ + w) * 3 + 0];
            const float th  = P[(l * NQ + w) * 3 + 1];
            const float om  = P[(l * NQ + w) * 3 + 2];
            const float ch = cosf(0.5f * th), sh = sinf(0.5f * th);
            const float aa = 0.5f * (phi + om), bb = 0.5f * (phi - om);
            const float ca = cosf(aa), sa = sinf(aa), cb = cosf(bb), sb = sinf(bb);
            /* m00 = (ch*ca, -ch*sa)  m01 = (-sh*cb, -sh*sb)
               m10 = (sh*cb, -sh*sb)  m11 = ( ch*ca,  ch*sa) */
            const float m00r =  ch * ca, m00i = -ch * sa;
            const float m01r = -sh * cb, m01i = -sh * sb;
            const float m10r =  sh * cb, m10i = -sh * sb;
            const float m11r =  ch * ca, m11i =  ch * sa;

            __syncthreads();
            sr0[k] = re0; si0[k] = im0; sr1[k] = re1; si1[k] = im1;
            __syncthreads();
            const int partner = k ^ (1 << p);
            const float pr0 = sr0[partner], pi0 = si0[partner];
            const float pr1 = sr1[partner], pi1 = si1[partner];
            if (((k >> p) & 1) == 0) {
                /* new0 = m00*mine + m01*partner */
                float nr0 = m00r * re0 - m00i * im0 + m01r * pr0 - m01i * pi0;
                float ni0 = m00r * im0 + m00i * re0 + m01r * pi0 + m01i * pr0;
                float nr1 = m00r * re1 - m00i * im1 + m01r * pr1 - m01i * pi1;
                float ni1 = m00r * im1 + m00i * re1 + m01r * pi1 + m01i * pr1;
                re0 = nr0; im0 = ni0; re1 = nr1; im1 = ni1;
            } else {
                /* new1 = m10*partner + m11*mine */
                float nr0 = m10r * pr0 - m10i * pi0 + m11r * re0 - m11i * im0;
                float ni0 = m10r * pi0 + m10i * pr0 + m11r * im0 + m11i * re0;
                float nr1 = m10r * pr1 - m10i * pi1 + m11r * re1 - m11i * im1;
                float ni1 = m10r * pi1 + m10i * pr1 + m11r * im1 + m11i * re1;
                re0 = nr0; im0 = ni0; re1 = nr1; im1 = ni1;
            }
        }
        /* CNOT chain: final[k] = v[sigma0(sigma1(...sigma6(k)))] */
        __syncthreads();
        sr0[k] = re0; si0[k] = im0; sr1[k] = re1; si1[k] = im1;
        __syncthreads();
        int j = k;
        for (int w = 6; w >= 0; --w)
            j ^= ((j >> (7 - w)) & 1) << (6 - w);
        re0 = sr0[j]; im0 = si0[j]; re1 = sr1[j]; im1 = si1[j];
    }

    /* partial sums: S00, S11, S01, and sum of hidden-part weights */
    const float z = (k < 128) ? 1.0f : -1.0f;
    const float p00 = z * (re0 * re0 + im0 * im0);
    const float p11 = z * (re1 * re1 + im1 * im1);
    const float p01 = z * (im0 * re1 - re0 * im1);   /* Im(col0 * conj(col1)) */
    const float pws = W[512 + k] + W[768 + k];        /* row 0, hidden half    */

    __syncthreads();
    sr0[k] = p00; si0[k] = p11; sr1[k] = p01; si1[k] = pws;
    __syncthreads();
    for (int s = 128; s > 0; s >>= 1) {
        if (k < s) {
            sr0[k] += sr0[k + s]; si0[k] += si0[k + s];
            sr1[k] += sr1[k + s]; si1[k] += si1[k + s];
        }
        __syncthreads();
    }
    if (k == 0) {
        const float S00 = sr0[0], S11 = si0[0], S01 = sr1[0];
        consts[g * 8 + 0] = 0.5f * (S00 + S11);   /* C0 */
        consts[g * 8 + 1] = 0.5f * (S00 - S11);   /* C1 */
        consts[g * 8 + 2] = -S01;                 /* C2 */
        consts[g * 8 + 3] = si1[0];               /* sum W[0,512:1024] */
        consts[g * 8 + 4] = bv[0];                /* bias */
    }
}

/* =====================================================================
 * Kernel 2: pack weight tile WB2[k2][n] = {W_n[2k2], W_n[2k2+1]}, n<4
 * (columns 4..15 zero) — matches the B-fragment lane layout.
 * ===================================================================== */
__global__ void qlstm_pack_wb(const float* __restrict__ Wf, const float* __restrict__ Wi,
                              const float* __restrict__ Wg, const float* __restrict__ Wo,
                              float* __restrict__ wb) {
    const int idx = blockIdx.x * blockDim.x + threadIdx.x;   /* 0..4095 */
    if (idx >= 256 * 16) return;
    const int k2 = idx >> 4;
    const int n  = idx & 15;
    float v0 = 0.0f, v1 = 0.0f;
    if (n < 4) {
        const float* W = (n == 0) ? Wf : (n == 1) ? Wi : (n == 2) ? Wg : Wo;
        v0 = W[2 * k2];       /* row 0 of W, x-part (cols 0..511) */
        v1 = W[2 * k2 + 1];
    }
    wb[idx * 2 + 0] = v0;
    wb[idx * 2 + 1] = v1;
}

/* =====================================================================
 * Kernel 3: projection GEMM (32768 x 512) @ (512 x 16) via
 * V_WMMA_F32_16X16X4_F32. One wave per 16-row tile, K streamed in 4s.
 * A frag (32-bit 16x4): lanes 0-15 hold K=0,1; lanes 16-31 hold K=2,3.
 * ===================================================================== */
__global__ void qlstm_proj_wmma(const float* __restrict__ X, const float* __restrict__ WB,
                                float* __restrict__ pre) {
    const int wave = (blockIdx.x * blockDim.x + threadIdx.x) >> 5;
    const int lane = threadIdx.x & 31;
    const int rowBase = wave * 16;
    const int m    = lane & 15;          /* A row within tile / B column n */
    const int koff = (lane >> 4) << 1;   /* 0 for lanes 0-15, 2 for 16-31  */

    const float* arow = X  + (size_t)(rowBase + m) * DFEAT + koff;
    const float* bcol = WB + (koff >> 1) * 32 + m * 2;

    v8f acc = {};
    for (int kb = 0; kb < DFEAT; kb += 4) {
        v2f a = *(const v2f*)(arow + kb);
        v2f b = *(const v2f*)(bcol + (kb >> 1) * 32);
        acc = __builtin_amdgcn_wmma_f32_16x16x4_f32(
            /*neg_a=*/false, a, /*neg_b=*/false, b,
            /*c_mod=*/(short)0, acc, /*reuse_a=*/false, /*reuse_b=*/false);
    }

    /* D layout: VGPR r -> M = r + 8*(lane>=16), N = lane&15. Keep N<4. */
    if (m < 4) {
        const int half = lane >> 4;
#pragma unroll
        for (int r = 0; r < 8; ++r) {
            const int row = rowBase + r + half * 8;
            pre[row * 4 + m] = acc[r];
        }
    }
}

/* =====================================================================
 * Kernel 4: scalar LSTM recurrence (hidden state is uniform across H,
 * so h,c collapse to per-batch scalars). 128 lanes, T=256 steps.
 * ===================================================================== */
__device__ __forceinline__ float sigmoidf_(float x) { return 1.0f / (1.0f + expf(-x)); }

__global__ void qlstm_recur(const float* __restrict__ pre, const float* __restrict__ consts,
                            float* __restrict__ hseq, float* __restrict__ hfin,
                            float* __restrict__ cfin) {
    const int b = threadIdx.x;           /* 0..127 */
    const float C0f = consts[0],  C1f = consts[1],  C2f = consts[2],  Sf = consts[3],  Bf = consts[4];
    const float C0i = consts[8],  C1i = consts[9],  C2i = consts[10], Si = consts[11], Bi = consts[12];
    const float C0g = consts[16], C1g = consts[17], C2g = consts[18], Sg = consts[19], Bg = consts[20];
    const float C0o = consts[24], C1o = consts[25], C2o = consts[26], So = consts[27], Bo = consts[28];

    float h = 0.0f, c = 0.0f;
    for (int t = 0; t < TSTEPS; ++t) {
        const float4 p4 = *(const float4*)(pre + (size_t)(t * BATCH + b) * 4);
        const float thf = p4.x + h * Sf + Bf;
        const float thi = p4.y + h * Si + Bi;
        const float thg = p4.z + h * Sg + Bg;
        const float tho = p4.w + h * So + Bo;
        const float fg = sigmoidf_(C0f + C1f * cosf(thf) + C2f * sinf(thf));
        const float ig = sigmoidf_(C0i + C1i * cosf(thi) + C2i * sinf(thi));
        const float gg = tanhf    (C0g + C1g * cosf(thg) + C2g * sinf(thg));
        const float og = sigmoidf_(C0o + C1o * cosf(tho) + C2o * sinf(tho));
        c = fg * c + ig * gg;
        h = og * tanhf(c);
        hseq[t * BATCH + b] = h;
    }
    hfin[b] = h;
    cfin[b] = c;
}

/* =====================================================================
 * Kernel 5: broadcast expand to d_out = [outputs(T,B,512) | hx | cx]
 * One block per logical row, 128 lanes x float4 = 512 floats.
 * ===================================================================== */
__global__ void qlstm_bcast(const float* __restrict__ hseq, const float* __restrict__ hfin,
                            const float* __restrict__ cfin, float* __restrict__ out) {
    const int row = blockIdx.x;          /* 0 .. 33023 */
    float v;
    if (row < NROWS)              v = hseq[row];
    else if (row < NROWS + BATCH) v = hfin[row - NROWS];
    else                          v = cfin[row - NROWS - BATCH];
    const float4 vv = make_float4(v, v, v, v);
    ((float4*)out)[(size_t)row * 128 + threadIdx.x] = vv;
}

/* ===================================================================== */
extern "C" void kernel_launch(void* const* d_in, const int* in_sizes, int n_in,
                              void* d_out, int out_size, void* d_ws, size_t ws_size,
                              hipStream_t stream) {
    const float* X  = (const float*)d_in[0];
    const float* Wf = (const float*)d_in[1];  const float* bf = (const float*)d_in[2];  const float* Pf = (const float*)d_in[3];
    const float* Wi = (const float*)d_in[4];  const float* bi = (const float*)d_in[5];  const float* Pi = (const float*)d_in[6];
    const float* Wg = (const float*)d_in[7];  const float* bg = (const float*)d_in[8];  const float* Pg = (const float*)d_in[9];
    const float* Wo = (const float*)d_in[10]; const float* bo = (const float*)d_in[11]; const float* Po = (const float*)d_in[12];

    char* ws = (char*)d_ws;
    float* consts = (float*)(ws + CONST_OFF);
    float* wb     = (float*)(ws + WB_OFF);
    float* pre    = (float*)(ws + PRE_OFF);
    float* hseq   = (float*)(ws + HSEQ_OFF);
    float* hfin   = (float*)(ws + HFIN_OFF);
    float* cfin   = (float*)(ws + CFIN_OFF);
    float* out    = (float*)d_out;

    qlstm_gate_consts<<<4, 256, 0, stream>>>(Wf, bf, Pf, Wi, bi, Pi, Wg, bg, Pg, Wo, bo, Po, consts);
    qlstm_pack_wb<<<16, 256, 0, stream>>>(Wf, Wi, Wg, Wo, wb);
    /* 2048 tiles, 8 waves (256 thr) per block -> 256 blocks */
    qlstm_proj_wmma<<<256, 256, 0, stream>>>(X, wb, pre);
    qlstm_recur<<<1, 128, 0, stream>>>(pre, consts, hseq, hfin, cfin);
    qlstm_bcast<<<NROWS + 2 * BATCH, 128, 0, stream>>>(hseq, hfin, cfin, out);
}